// IntersectLoss_5557687681378
// MI455X (gfx1250) — compile-verified
//
#include <hip/hip_runtime.h>

#define BLK 256
#define GX 32   // blocks along pair dimension
#define GY 32   // batch dimension (B)
#define NPART (GX * GY)

typedef __attribute__((ext_vector_type(2))) float v2f;
typedef __attribute__((ext_vector_type(8))) float v8f;

struct F3 { float x, y, z; };

__device__ __forceinline__ F3 f3sub(const F3& a, const F3& b) {
    return F3{a.x - b.x, a.y - b.y, a.z - b.z};
}
__device__ __forceinline__ F3 f3cross(const F3& a, const F3& b) {
    return F3{a.y * b.z - a.z * b.y,
              a.z * b.x - a.x * b.z,
              a.x * b.y - a.y * b.x};
}
__device__ __forceinline__ float f3dot(const F3& a, const F3& b) {
    return a.x * b.x + a.y * b.y + a.z * b.z;
}

// Conic distance-field penetration: cone tri C (centroid+normal field),
// query tri Q (3 verts tested against field). Matches _cone_point2plane.
__device__ __forceinline__ float cone_pen(const F3* C, const F3* Q) {
    const float third = 1.0f / 3.0f;
    F3 c = F3{(C[0].x + C[1].x + C[2].x) * third,
              (C[0].y + C[1].y + C[2].y) * third,
              (C[0].z + C[1].z + C[2].z) * third};
    F3 e1 = f3sub(C[1], C[0]);
    F3 e2 = f3sub(C[2], C[0]);
    F3 n  = f3cross(e1, e2);
    float inv = 1.0f / (sqrtf(f3dot(n, n)) + 1e-8f);
    n.x *= inv; n.y *= inv; n.z *= inv;

    float pen = 0.0f;
#pragma unroll
    for (int v = 0; v < 3; ++v) {
        F3 d = f3sub(Q[v], c);
        float h   = f3dot(d, n);              // signed plane distance
        float r   = sqrtf(f3dot(d, d));       // distance to centroid
        float phi = fmaxf(0.5f - r, 0.0f);    // SIGMA = 0.5
        pen += (h < 0.0f) ? phi * h * h : 0.0f;
    }
    return pen;
}

__device__ __forceinline__ void load_tri(const float* __restrict__ vb,
                                         const int* __restrict__ faces,
                                         int f, F3* T) {
    const int* fp = faces + 3 * f;
#pragma unroll
    for (int k = 0; k < 3; ++k) {
        const float* vp = vb + (size_t)fp[k] * 3;
        T[k] = F3{vp[0], vp[1], vp[2]};
    }
}

// Stage 1: stream collision pairs, gather triangles, accumulate penetration.
// Deterministic: fixed grid-stride order per thread, fixed LDS tree reduce.
__global__ void __launch_bounds__(BLK)
pair_kernel(const float* __restrict__ vertices,   // [B, NV, 3]
            const int*   __restrict__ faces,      // [NF, 3]
            const int*   __restrict__ coll,       // [B, NC, 2]
            int NV, int NC,
            float* __restrict__ partials) {
    __shared__ float sdata[BLK];
    const int tid = threadIdx.x;
    const int b   = blockIdx.y;

    const int*   cb = coll     + (size_t)b * NC * 2;
    const float* vb = vertices + (size_t)b * NV * 3;

    const int stride = gridDim.x * BLK;
    float acc = 0.0f;

    for (int pc = blockIdx.x * BLK + tid; pc < NC; pc += stride) {
        // CDNA5: speculative prefetch of next chunk of the index stream
        // (global_prefetch_b8; translation failures past the end are dropped).
        __builtin_prefetch(cb + 2 * (pc + stride), 0, 0);

        int2 pr = ((const int2*)cb)[pc];
        bool valid = pr.x >= 0;                 // -1 padding slots
        int rc = max(pr.x, 0);
        int ic = max(pr.y, 0);

        F3 R[3], I[3];
        load_tri(vb, faces, rc, R);
        load_tri(vb, faces, ic, I);

        float pen = cone_pen(R, I) + cone_pen(I, R);
        acc += valid ? pen : 0.0f;
    }

    sdata[tid] = acc;
    __syncthreads();
#pragma unroll
    for (int s = BLK / 2; s > 0; s >>= 1) {
        if (tid < s) sdata[tid] += sdata[tid + s];
        __syncthreads();
    }
    if (tid == 0) partials[blockIdx.y * gridDim.x + blockIdx.x] = sdata[0];
}

// Stage 2: deterministic final reduction of NPART partials.
// The 32->16 lane fold uses V_WMMA_F32_16X16X4_F32 with an all-ones A matrix:
// D[m,n] = sum_k B[k,n], so with lane partials in B's first VGPR (second = 0)
// every D entry is a column sum; summing D[0, 0..15] (VGPR0, lanes 0-15) gives
// the total. This is layout-independent w.r.t. B's K-row placement.
__global__ void __launch_bounds__(BLK)
reduce_kernel(const float* __restrict__ partials,
              const float* __restrict__ weight,
              float* __restrict__ out,
              float invB) {
    __shared__ float sdata[BLK];
    const int tid = threadIdx.x;

    float a = partials[tid] + partials[tid + 256] +
              partials[tid + 512] + partials[tid + 768];
    sdata[tid] = a;
    __syncthreads();

    if (tid < 32) {                 // wave 0 only: EXEC all ones for WMMA
        float acc = 0.0f;
#pragma unroll
        for (int k = 0; k < 8; ++k) acc += sdata[tid + 32 * k];

        v2f am; am[0] = 1.0f; am[1] = 1.0f;   // A = ones (16x4)
        v2f bm; bm[0] = acc;  bm[1] = 0.0f;   // B row data = lane partials
        v8f cm = {};
        cm = __builtin_amdgcn_wmma_f32_16x16x4_f32(
                /*neg_a=*/false, am, /*neg_b=*/false, bm,
                /*c_mod=*/(short)0, cm, /*reuse_a=*/false, /*reuse_b=*/false);

        float s = cm[0];                       // D[0, lane] for lanes 0..15
        s += __shfl_xor(s, 1, 32);
        s += __shfl_xor(s, 2, 32);
        s += __shfl_xor(s, 4, 32);
        s += __shfl_xor(s, 8, 32);
        if (tid == 0) out[0] = s * invB * weight[0];
    }
}

extern "C" void kernel_launch(void* const* d_in, const int* in_sizes, int n_in,
                              void* d_out, int out_size, void* d_ws, size_t ws_size,
                              hipStream_t stream) {
    // inputs: 0 pose, 1 joints, 2 points, 3 keypoints, 4 vertices,
    //         5 weight, 6 faces, 7 collision_idxs
    const float* vertices = (const float*)d_in[4];
    const float* weight   = (const float*)d_in[5];
    const int*   faces    = (const int*)d_in[6];
    const int*   coll     = (const int*)d_in[7];

    const int B  = GY;                            // 32
    const int NV = in_sizes[4] / (B * 3);         // 6890
    const int NC = (in_sizes[7] / 2) / B;         // NF * MAXC

    float* partials = (float*)d_ws;               // NPART floats (4 KB)

    dim3 grid(GX, GY);
    pair_kernel<<<grid, BLK, 0, stream>>>(vertices, faces, coll, NV, NC, partials);
    reduce_kernel<<<1, BLK, 0, stream>>>(partials, weight, (float*)d_out, 1.0f / B);
}